// PNA_77360950936125
// MI455X (gfx1250) — compile-verified
//
#include <hip/hip_runtime.h>
#include <math.h>

#define NN 131072
#define NE 524288
#define NG 4096
#define AVGD 1.3040748974f

typedef float v2f __attribute__((ext_vector_type(2)));
typedef float v8f __attribute__((ext_vector_type(8)));

__device__ __forceinline__ unsigned fenc(float f) {
  unsigned u = __float_as_uint(f);
  return (u & 0x80000000u) ? ~u : (u | 0x80000000u);
}
__device__ __forceinline__ float fdec(unsigned e) {
  unsigned u = (e & 0x80000000u) ? (e & 0x7fffffffu) : ~e;
  return __uint_as_float(u);
}
__device__ __forceinline__ float sigm(float x) { return 1.f / (1.f + expf(-x)); }

// ---------------- lin1: h = relu(x @ W.T + b), [N,32] -> [N,64] -------------
__global__ void __launch_bounds__(256) k_lin1(const float* __restrict__ x,
                                              const float* __restrict__ W,
                                              const float* __restrict__ b,
                                              float* __restrict__ h) {
  int lane = threadIdx.x & 31, wv = threadIdx.x >> 5;
  int tb = (blockIdx.x * 8 + wv) * 16;
  int m = lane & 15, kb = (lane >> 4) * 2, nn = lane & 15, rb = (lane >> 4) * 8;
  v8f acc[4];
#pragma unroll
  for (int t = 0; t < 4; ++t) {
    float bv = b[t * 16 + nn];
#pragma unroll
    for (int r = 0; r < 8; ++r) acc[t][r] = bv;
  }
#pragma unroll
  for (int kk = 0; kk < 32; kk += 4) {
    float2 av = *(const float2*)(x + (tb + m) * 32 + kk + kb);
    v2f a; a.x = av.x; a.y = av.y;
#pragma unroll
    for (int t = 0; t < 4; ++t) {
      int n = t * 16 + nn;
      float2 bv = *(const float2*)(W + n * 32 + kk + kb);
      v2f bb; bb.x = bv.x; bb.y = bv.y;
      acc[t] = __builtin_amdgcn_wmma_f32_16x16x4_f32(false, a, false, bb, (short)0,
                                                     acc[t], false, false);
    }
  }
#pragma unroll
  for (int t = 0; t < 4; ++t) {
    int n = t * 16 + nn;
#pragma unroll
    for (int r = 0; r < 8; ++r) h[(tb + rb + r) * 64 + n] = fmaxf(acc[t][r], 0.f);
  }
}

// ------------- per-layer weight fusion: Wm[64x136], Wp[64x832] --------------
__global__ void k_prep(const float* __restrict__ preW, const float* __restrict__ preb,
                       const float* __restrict__ eW, const float* __restrict__ eb,
                       const float* __restrict__ postW, const float* __restrict__ postb,
                       const float* __restrict__ clW, const float* __restrict__ clb,
                       float* Wm, float* bm, float* Wp, float* bp) {
  int tid = threadIdx.x;
  for (int i = tid; i < 64 * 136; i += 256) {
    int o = i / 136, k = i - o * 136;
    float v = 0.f;
    if (k < 128) v = preW[o * 192 + k];
    else if (k < 134) {
      int j = k - 128;
      for (int hh = 0; hh < 64; ++hh) v = fmaf(preW[o * 192 + 128 + hh], eW[hh * 6 + j], v);
    }
    Wm[i] = v;
  }
  for (int i = tid; i < 64; i += 256) {
    float v = preb[i];
    for (int hh = 0; hh < 64; ++hh) v = fmaf(preW[i * 192 + 128 + hh], eb[hh], v);
    bm[i] = v;
  }
  for (int i = tid; i < 64 * 832; i += 256) {
    int o = i / 832, k = i - o * 832;
    float v = 0.f;
    for (int hh = 0; hh < 64; ++hh) v = fmaf(clW[o * 64 + hh], postW[hh * 832 + k], v);
    Wp[i] = v;
  }
  for (int i = tid; i < 64; i += 256) {
    float v = clb[i];
    for (int hh = 0; hh < 64; ++hh) v = fmaf(clW[i * 64 + hh], postb[hh], v);
    bp[i] = v;
  }
}

// ------------- init aggregation buffers per layer ---------------------------
__global__ void k_init(float* sum, float* sumsq, unsigned* mnb, unsigned* mxb,
                       float* cnt, float* stat) {
  int i = blockIdx.x * 256 + threadIdx.x;
  if (i < NN * 64) {
    sum[i] = 0.f; sumsq[i] = 0.f;
    mnb[i] = fenc(INFINITY); mxb[i] = fenc(-INFINITY);
  }
  if (i < NN) cnt[i] = 0.f;
  if (i < 128) stat[i] = 0.f;
}

// ------------- edge messages (K=136 WMMA) + atomic segment aggregation ------
// Gather of x[dst]/x[src] rows uses GLOBAL_LOAD_ASYNC_TO_LDS_B128:
// lanes 0-15 fetch the dst row (16B each), lanes 16-31 the src row,
// written straight into this wave's LDS tile (16B-aligned, stride 140 floats).
__global__ void __launch_bounds__(256) k_edge(const float* __restrict__ h,
                                              const float* __restrict__ ea,
                                              const int* __restrict__ eidx,
                                              const float* __restrict__ Wm,
                                              const float* __restrict__ bm,
                                              float* sum, float* sumsq,
                                              unsigned* mnb, unsigned* mxb, float* cnt) {
  __shared__ float lin[8][16 * 140];
  __shared__ int ldst[8][16];
  int lane = threadIdx.x & 31, wv = threadIdx.x >> 5;
  int tb = (blockIdx.x * 8 + wv) * 16;
  int half = lane >> 4;   // 0: dst row, 1: src row
  int l16 = lane & 15;
  for (int e = 0; e < 16; ++e) {
    int eg = tb + e;
    int dn = eidx[NE + eg];
    int sn = eidx[eg];
    float* row = &lin[wv][e * 140];
    const float* gsrc = (half ? (h + sn * 64) : (h + dn * 64)) + l16 * 4;
    // LDS byte offset = low 32 bits of the generic pointer (ISA aperture rule)
    unsigned ldsoff = (unsigned)(uintptr_t)row + (unsigned)(half * 256 + l16 * 16);
    asm volatile("global_load_async_to_lds_b128 %0, %1, off"
                 :: "v"(ldsoff), "v"(gsrc) : "memory");
    if (lane < 8) row[128 + lane] = (lane < 6) ? ea[eg * 6 + lane] : 0.f;
    if (lane == 0) ldst[wv][e] = dn;
  }
  asm volatile("s_wait_asynccnt 0x0" ::: "memory");
  __syncthreads();
  int m = lane & 15, kb = (lane >> 4) * 2, nn = lane & 15, rb = (lane >> 4) * 8;
  const float* arow = &lin[wv][m * 140];
  v8f acc[4];
#pragma unroll
  for (int t = 0; t < 4; ++t) {
    float bv = bm[t * 16 + nn];
#pragma unroll
    for (int r = 0; r < 8; ++r) acc[t][r] = bv;
  }
  for (int kk = 0; kk < 136; kk += 4) {
    float2 av = *(const float2*)(arow + kk + kb);
    v2f a; a.x = av.x; a.y = av.y;
#pragma unroll
    for (int t = 0; t < 4; ++t) {
      float2 bv = *(const float2*)(Wm + (t * 16 + nn) * 136 + kk + kb);
      v2f bb; bb.x = bv.x; bb.y = bv.y;
      acc[t] = __builtin_amdgcn_wmma_f32_16x16x4_f32(false, a, false, bb, (short)0,
                                                     acc[t], false, false);
    }
  }
#pragma unroll
  for (int t = 0; t < 4; ++t) {
    int n = t * 16 + nn;
#pragma unroll
    for (int r = 0; r < 8; ++r) {
      int d = ldst[wv][rb + r];
      float v = acc[t][r];
      int g = d * 64 + n;
      atomicAdd(&sum[g], v);
      atomicAdd(&sumsq[g], v * v);
      unsigned k = fenc(v);
      atomicMin(&mnb[g], k);
      atomicMax(&mxb[g], k);
    }
  }
  if (lane < 16) atomicAdd(&cnt[ldst[wv][lane]], 1.f);
}

// ------------- node post: build [x|agg|agg*s1|agg*s2], K=832 WMMA -----------
__global__ void __launch_bounds__(128) k_post(const float* __restrict__ hin,
                                              const float* __restrict__ sum,
                                              const float* __restrict__ sumsq,
                                              const unsigned* __restrict__ mnb,
                                              const unsigned* __restrict__ mxb,
                                              const float* __restrict__ cnt,
                                              const float* __restrict__ Wp,
                                              const float* __restrict__ bp,
                                              float* __restrict__ hout) {
  __shared__ float lin[4][16 * 322];
  __shared__ float lw[64 * 66];
  __shared__ float lscal[4][16][2];
  int tid = threadIdx.x, lane = tid & 31, wv = tid >> 5;
  int nb = (blockIdx.x * 4 + wv) * 16;
  for (int nd = 0; nd < 16; ++nd) {
    int node = nb + nd;
    float* row = &lin[wv][nd * 322];
    float c0 = cnt[node];
    float c = fmaxf(c0, 1.f);
    bool has = c0 > 0.f;
    for (int f = lane; f < 64; f += 32) {
      int g = node * 64 + f;
      float me = sum[g] / c, me2 = sumsq[g] / c;
      row[f] = hin[g];
      row[64 + f] = me;
      row[128 + f] = has ? fdec(mnb[g]) : 0.f;
      row[192 + f] = has ? fdec(mxb[g]) : 0.f;
      row[256 + f] = sqrtf(fmaxf(me2 - me * me, 0.f) + 1e-5f);
    }
    if (lane == 0) {
      float logd = logf(c + 1.f);
      lscal[wv][nd][0] = logd / AVGD;
      lscal[wv][nd][1] = AVGD / logd;
    }
  }
  __syncthreads();
  int m = lane & 15, kb = (lane >> 4) * 2, nn = lane & 15, rb = (lane >> 4) * 8;
  float s1 = lscal[wv][m][0], s2 = lscal[wv][m][1];
  const float* arow = &lin[wv][m * 322];
  v8f acc[4];
#pragma unroll
  for (int t = 0; t < 4; ++t) {
    float bv = bp[t * 16 + nn];
#pragma unroll
    for (int r = 0; r < 8; ++r) acc[t][r] = bv;
  }
  for (int kc = 0; kc < 832; kc += 64) {
    __syncthreads();
    for (int i = tid; i < 64 * 64; i += 128) {
      int n = i >> 6, k = i & 63;
      lw[n * 66 + k] = Wp[n * 832 + kc + k];
    }
    __syncthreads();
    int off = (kc < 320) ? 0 : ((kc < 576) ? 256 : 512);
    float sm = (kc < 320) ? 1.f : ((kc < 576) ? s1 : s2);
    for (int kk = 0; kk < 64; kk += 4) {
      int k = kc + kk + kb;
      float2 av = *(const float2*)(arow + (k - off));
      v2f a; a.x = av.x * sm; a.y = av.y * sm;
#pragma unroll
      for (int t = 0; t < 4; ++t) {
        float2 bv = *(const float2*)(lw + (t * 16 + nn) * 66 + kk + kb);
        v2f bb; bb.x = bv.x; bb.y = bv.y;
        acc[t] = __builtin_amdgcn_wmma_f32_16x16x4_f32(false, a, false, bb, (short)0,
                                                       acc[t], false, false);
      }
    }
  }
#pragma unroll
  for (int t = 0; t < 4; ++t) {
    int n = t * 16 + nn;
#pragma unroll
    for (int r = 0; r < 8; ++r) hout[(nb + rb + r) * 64 + n] = acc[t][r];
  }
}

// ------------- batch-norm stats + apply -------------------------------------
__global__ void k_bnstats(const float* __restrict__ ht, float* stat) {
  __shared__ float ls[256], lq[256];
  int tid = threadIdx.x;
  int f = tid & 63, grp = tid >> 6;
  float s = 0.f, q = 0.f;
  for (int r = blockIdx.x * 4 + grp; r < NN; r += gridDim.x * 4) {
    float v = ht[r * 64 + f];
    s += v; q = fmaf(v, v, q);
  }
  ls[tid] = s; lq[tid] = q;
  __syncthreads();
  if (tid < 64) {
    float a = ls[tid] + ls[64 + tid] + ls[128 + tid] + ls[192 + tid];
    float b = lq[tid] + lq[64 + tid] + lq[128 + tid] + lq[192 + tid];
    atomicAdd(&stat[tid], a);
    atomicAdd(&stat[64 + tid], b);
  }
}

__global__ void k_bnapply(const float* __restrict__ ht, const float* __restrict__ stat,
                          const float* __restrict__ g, const float* __restrict__ b,
                          float* __restrict__ h) {
  int i = blockIdx.x * 256 + threadIdx.x;
  if (i >= NN * 64) return;
  int f = i & 63;
  float mean = stat[f] * (1.f / NN);
  float var = stat[64 + f] * (1.f / NN) - mean * mean;
  float v = (ht[i] - mean) * rsqrtf(var + 1e-5f) * g[f] + b[f];
  h[i] = fmaxf(v, 0.f);
}

// ------------- set2set -------------------------------------------------------
__global__ void k_s2s_init(float* qst, float* hsA, float* cs) {
  int i = blockIdx.x * 256 + threadIdx.x;
  if (i < NG * 128) qst[i] = 0.f;
  if (i < NG * 64) { hsA[i] = 0.f; cs[i] = 0.f; }
}

__global__ void k_lstm(const float* __restrict__ qst, const float* __restrict__ hprev,
                       float* __restrict__ hnew, float* __restrict__ cs,
                       const float* __restrict__ Wih, const float* __restrict__ Whh,
                       const float* __restrict__ bih, const float* __restrict__ bhh) {
  int idx = blockIdx.x * 256 + threadIdx.x;
  if (idx >= NG * 64) return;
  int g = idx >> 6, k = idx & 63;
  float gi = bih[k] + bhh[k];
  float gf = bih[64 + k] + bhh[64 + k];
  float gg = bih[128 + k] + bhh[128 + k];
  float go = bih[192 + k] + bhh[192 + k];
  const float* q = qst + g * 128;
  for (int j = 0; j < 128; ++j) {
    float qv = q[j];
    gi = fmaf(Wih[k * 128 + j], qv, gi);
    gf = fmaf(Wih[(64 + k) * 128 + j], qv, gf);
    gg = fmaf(Wih[(128 + k) * 128 + j], qv, gg);
    go = fmaf(Wih[(192 + k) * 128 + j], qv, go);
  }
  const float* hp = hprev + g * 64;
  for (int j = 0; j < 64; ++j) {
    float hv = hp[j];
    gi = fmaf(Whh[k * 64 + j], hv, gi);
    gf = fmaf(Whh[(64 + k) * 64 + j], hv, gf);
    gg = fmaf(Whh[(128 + k) * 64 + j], hv, gg);
    go = fmaf(Whh[(192 + k) * 64 + j], hv, go);
  }
  float c = sigm(gf) * cs[idx] + sigm(gi) * tanhf(gg);
  cs[idx] = c;
  hnew[idx] = sigm(go) * tanhf(c);
}

__global__ void k_att_init(float* r, unsigned* emx, float* es) {
  int i = blockIdx.x * 256 + threadIdx.x;
  if (i < NG * 64) r[i] = 0.f;
  if (i < NG) { emx[i] = fenc(-INFINITY); es[i] = 0.f; }
}

__global__ void k_e(const float* __restrict__ h, const float* __restrict__ hs,
                    const int* __restrict__ batch, float* ev, unsigned* emx) {
  int n = blockIdx.x * 256 + threadIdx.x;
  if (n >= NN) return;
  int g = batch[n];
  const float* hr = h + n * 64;
  const float* hg = hs + g * 64;
  float e = 0.f;
  for (int j = 0; j < 64; ++j) e = fmaf(hr[j], hg[j], e);
  ev[n] = e;
  atomicMax(&emx[g], fenc(e));
}

__global__ void k_ex(const float* __restrict__ ev, const int* __restrict__ batch,
                     const unsigned* __restrict__ emx, float* exv, float* es) {
  int n = blockIdx.x * 256 + threadIdx.x;
  if (n >= NN) return;
  int g = batch[n];
  float em = fdec(emx[g]);
  if (!(em > -3.0e38f)) em = 0.f;  // reference: non-finite segment max -> 0
  float x = expf(ev[n] - em);
  exv[n] = x;
  atomicAdd(&es[g], x);
}

__global__ void k_r(const float* __restrict__ h, const float* __restrict__ exv,
                    const float* __restrict__ es, const int* __restrict__ batch,
                    float* r) {
  int i = blockIdx.x * 256 + threadIdx.x;
  if (i >= NN * 64) return;
  int n = i >> 6, k = i & 63;
  int g = batch[n];
  float a = exv[n] / (es[g] + 1e-16f);
  atomicAdd(&r[g * 64 + k], a * h[i]);
}

__global__ void k_qstar(const float* __restrict__ hs, const float* __restrict__ r,
                        float* qst) {
  int i = blockIdx.x * 256 + threadIdx.x;
  if (i >= NG * 64) return;
  int g = i >> 6, k = i & 63;
  qst[g * 128 + k] = hs[i];
  qst[g * 128 + 64 + k] = r[i];
}

// ------------- final MLP -----------------------------------------------------
__global__ void k_mlp1(const float* __restrict__ qst, const float* __restrict__ t,
                       const float* __restrict__ p, const float* __restrict__ W1,
                       const float* __restrict__ b1, float* m1) {
  int i = blockIdx.x * 256 + threadIdx.x;
  if (i >= NG * 64) return;
  int g = i >> 6, k = i & 63;
  const float* w = W1 + k * 130;
  const float* q = qst + g * 128;
  float v = b1[k];
  for (int j = 0; j < 128; ++j) v = fmaf(w[j], q[j], v);
  v = fmaf(w[128], t[g], v);
  v = fmaf(w[129], p[g], v);
  m1[i] = fmaxf(v, 0.f);
}

__global__ void k_mlp2(const float* __restrict__ m1, const float* __restrict__ W2,
                       const float* __restrict__ b2, float* m2) {
  int i = blockIdx.x * 256 + threadIdx.x;
  if (i >= NG * 32) return;
  int g = i >> 5, k = i & 31;
  float v = b2[k];
  for (int j = 0; j < 64; ++j) v = fmaf(W2[k * 64 + j], m1[g * 64 + j], v);
  m2[i] = fmaxf(v, 0.f);
}

__global__ void k_mlp3(const float* __restrict__ m2, const float* __restrict__ W3,
                       const float* __restrict__ b3, float* out) {
  int g = blockIdx.x * 256 + threadIdx.x;
  if (g >= NG) return;
  float v = b3[0];
  for (int j = 0; j < 32; ++j) v = fmaf(W3[j], m2[g * 32 + j], v);
  out[g] = v;
}

// ---------------------------------------------------------------------------
extern "C" void kernel_launch(void* const* d_in, const int* in_sizes, int n_in,
                              void* d_out, int out_size, void* d_ws, size_t ws_size,
                              hipStream_t stream) {
  (void)in_sizes; (void)n_in; (void)out_size; (void)ws_size;
  const float* x     = (const float*)d_in[0];
  const float* ea    = (const float*)d_in[1];
  const float* t     = (const float*)d_in[2];
  const float* p     = (const float*)d_in[3];
  const float* lin1W = (const float*)d_in[4];
  const float* lin1b = (const float*)d_in[5];
  const float* edgeW = (const float*)d_in[6];
  const float* edgeb = (const float*)d_in[7];
  const float* preW  = (const float*)d_in[8];
  const float* preb  = (const float*)d_in[9];
  const float* postW = (const float*)d_in[10];
  const float* postb = (const float*)d_in[11];
  const float* clW   = (const float*)d_in[12];
  const float* clb   = (const float*)d_in[13];
  const float* bng   = (const float*)d_in[14];
  const float* bnb   = (const float*)d_in[15];
  const float* Wih   = (const float*)d_in[16];
  const float* Whh   = (const float*)d_in[17];
  const float* bih   = (const float*)d_in[18];
  const float* bhh   = (const float*)d_in[19];
  const float* W1    = (const float*)d_in[20];
  const float* b1    = (const float*)d_in[21];
  const float* W2    = (const float*)d_in[22];
  const float* b2    = (const float*)d_in[23];
  const float* W3    = (const float*)d_in[24];
  const float* b3    = (const float*)d_in[25];
  const int*   eidx  = (const int*)d_in[26];
  const int*   batch = (const int*)d_in[27];
  float* out = (float*)d_out;

  char* wsp = (char*)d_ws;
  auto alloc = [&](size_t bytes) -> char* {
    char* r = wsp;
    wsp += (bytes + 255) & ~(size_t)255;
    return r;
  };
  float*    h    = (float*)alloc((size_t)NN * 64 * 4);
  float*    htmp = (float*)alloc((size_t)NN * 64 * 4);
  float*    sumb = (float*)alloc((size_t)NN * 64 * 4);
  float*    sqb  = (float*)alloc((size_t)NN * 64 * 4);
  unsigned* mnb  = (unsigned*)alloc((size_t)NN * 64 * 4);
  unsigned* mxb  = (unsigned*)alloc((size_t)NN * 64 * 4);
  float*    cntb = (float*)alloc((size_t)NN * 4);
  float*    Wm   = (float*)alloc(64 * 136 * 4);
  float*    bm   = (float*)alloc(64 * 4);
  float*    Wp   = (float*)alloc(64 * 832 * 4);
  float*    bp   = (float*)alloc(64 * 4);
  float*    stat = (float*)alloc(128 * 4);
  float*    qst  = (float*)alloc((size_t)NG * 128 * 4);
  float*    hsA  = (float*)alloc((size_t)NG * 64 * 4);
  float*    hsB  = (float*)alloc((size_t)NG * 64 * 4);
  float*    csb  = (float*)alloc((size_t)NG * 64 * 4);
  float*    rb_  = (float*)alloc((size_t)NG * 64 * 4);
  float*    ev   = (float*)alloc((size_t)NN * 4);
  float*    exv  = (float*)alloc((size_t)NN * 4);
  unsigned* emx  = (unsigned*)alloc((size_t)NG * 4);
  float*    esum = (float*)alloc((size_t)NG * 4);
  float*    m1   = (float*)alloc((size_t)NG * 64 * 4);
  float*    m2   = (float*)alloc((size_t)NG * 32 * 4);

  k_lin1<<<NN / 128, 256, 0, stream>>>(x, lin1W, lin1b, h);

  for (int i = 0; i < 3; ++i) {
    k_prep<<<1, 256, 0, stream>>>(preW + i * 64 * 192, preb + i * 64,
                                  edgeW + i * 64 * 6, edgeb + i * 64,
                                  postW + i * 64 * 832, postb + i * 64,
                                  clW + i * 64 * 64, clb + i * 64,
                                  Wm, bm, Wp, bp);
    k_init<<<(NN * 64) / 256, 256, 0, stream>>>(sumb, sqb, mnb, mxb, cntb, stat);
    k_edge<<<NE / 128, 256, 0, stream>>>(h, ea, eidx, Wm, bm, sumb, sqb, mnb, mxb, cntb);
    k_post<<<NN / 64, 128, 0, stream>>>(h, sumb, sqb, mnb, mxb, cntb, Wp, bp, htmp);
    k_bnstats<<<1024, 256, 0, stream>>>(htmp, stat);
    k_bnapply<<<(NN * 64) / 256, 256, 0, stream>>>(htmp, stat, bng + i * 64, bnb + i * 64, h);
  }

  k_s2s_init<<<(NG * 128) / 256, 256, 0, stream>>>(qst, hsA, csb);
  float* hprev = hsA;
  float* hcur = hsB;
  for (int s = 0; s < 3; ++s) {
    k_lstm<<<(NG * 64) / 256, 256, 0, stream>>>(qst, hprev, hcur, csb, Wih, Whh, bih, bhh);
    k_att_init<<<(NG * 64) / 256, 256, 0, stream>>>(rb_, emx, esum);
    k_e<<<NN / 256, 256, 0, stream>>>(h, hcur, batch, ev, emx);
    k_ex<<<NN / 256, 256, 0, stream>>>(ev, batch, emx, exv, esum);
    k_r<<<(NN * 64) / 256, 256, 0, stream>>>(h, exv, esum, batch, rb_);
    k_qstar<<<(NG * 64) / 256, 256, 0, stream>>>(hcur, rb_, qst);
    float* tmp = hprev; hprev = hcur; hcur = tmp;
  }

  k_mlp1<<<(NG * 64) / 256, 256, 0, stream>>>(qst, t, p, W1, b1, m1);
  k_mlp2<<<(NG * 32) / 256, 256, 0, stream>>>(m1, W2, b2, m2);
  k_mlp3<<<NG / 256, 256, 0, stream>>>(m2, W3, b3, out);
}